// NCA_91010357002481
// MI455X (gfx1250) — compile-verified
//
#include <hip/hip_runtime.h>

typedef float v4f __attribute__((ext_vector_type(4)));
typedef int   v4i __attribute__((vector_size(16)));          // matches builtin's V4i
typedef __attribute__((address_space(1))) v4i* gv4i_p;       // global int4*
typedef __attribute__((address_space(3))) v4i* lv4i_p;       // LDS int4*

#define HH 1024
#define WW 1024
#define TILE_H 8
#define LROWS (TILE_H + 2)
#define TPB 256  // 256 threads * 4 floats = 1024 columns (one full row per sweep)

// ---- CDNA5 async global->LDS copy (ASYNCcnt-tracked) ----
__device__ __forceinline__ void async_ldg_to_lds_f4(const float* g, float* l) {
#if __has_builtin(__builtin_amdgcn_global_load_async_to_lds_b128)
    __builtin_amdgcn_global_load_async_to_lds_b128(
        (gv4i_p)(g), (lv4i_p)(l), /*imm offset*/ 0, /*cpol*/ 0);
#else
    *(v4f*)l = *(const v4f*)g;  // synchronous fallback
#endif
}

__device__ __forceinline__ void wait_async_lds() {
#if __has_builtin(__builtin_amdgcn_s_wait_asynccnt)
    __builtin_amdgcn_s_wait_asynccnt(0);
#else
    asm volatile("s_wait_asynccnt 0" ::: "memory");
#endif
}

__global__ __launch_bounds__(TPB) void nca_step_kernel(
    const float* __restrict__ hp, const float* __restrict__ wp,
    const float* __restrict__ bp, const float* __restrict__ dxp,
    const float* __restrict__ dyp, float* __restrict__ out, int Bn)
{
    extern __shared__ float smem[];
    float* const sF0 = smem;                   // h tile (LROWS x WW)
    float* const sF1 = smem + LROWS * WW;      // w tile
    float* const sF2 = smem + 2 * LROWS * WW;  // b tile

    const int tilesPerImg = HH / TILE_H;
    const int img = blockIdx.x / tilesPerImg;
    const int ty  = blockIdx.x - img * tilesPerImg;
    const int y0  = ty * TILE_H;
    const int x0  = (int)threadIdx.x * 4;

    const size_t plane   = (size_t)HH * WW;
    const size_t imgBase = (size_t)img * plane;
    const size_t tileBase = imgBase + (size_t)y0 * WW + x0;

    // ---- Stage h/w/b tile rows (clamped row index == replicate vertical halo)
    {
        const float* const gF[3] = { hp, wp, bp };
        float* const       lF[3] = { sF0, sF1, sF2 };
#pragma unroll
        for (int f = 0; f < 3; ++f) {
            const float* src = gF[f] + imgBase + x0;
            float*       dst = lF[f] + x0;
#pragma unroll
            for (int rr = 0; rr < LROWS; ++rr) {
                int gy = y0 - 1 + rr;
                gy = gy < 0 ? 0 : (gy > HH - 1 ? HH - 1 : gy);
                async_ldg_to_lds_f4(src + (size_t)gy * WW, dst + rr * WW);
            }
        }
    }

    // ---- Overlap: issue the single-use dem loads (nontemporal) while the
    //      async DMA is in flight; they don't depend on LDS.
    v4f dxv[TILE_H], dyv[TILE_H];
#pragma unroll
    for (int r = 0; r < TILE_H; ++r) {
        dxv[r] = __builtin_nontemporal_load((const v4f*)(dxp + tileBase + (size_t)r * WW));
        dyv[r] = __builtin_nontemporal_load((const v4f*)(dyp + tileBase + (size_t)r * WW));
    }

    wait_async_lds();
    __syncthreads();

    // Horizontal neighbor scalar indices with replicate clamp
    const int xl = (x0 > 0) ? (x0 - 1) : 0;
    const int xr = (x0 + 4 < WW) ? (x0 + 4) : (WW - 1);

    const size_t outPlane = (size_t)Bn * plane;
    float* const outH = out;
    float* const outW = out + outPlane;
    float* const outB = out + 2 * outPlane;

#pragma unroll
    for (int r = 0; r < TILE_H; ++r) {
        const int gy   = y0 + r;
        const int rowm = r * WW;
        const int rowc = rowm + WW;
        const int rowp = rowc + WW;

        // ---- h ----
        v4f hc = *(const v4f*)(sF0 + rowc + x0);
        v4f hu = *(const v4f*)(sF0 + rowm + x0);
        v4f hd = *(const v4f*)(sF0 + rowp + x0);
        float hls = sF0[rowc + xl], hrs = sF0[rowc + xr];
        // ---- w ----
        v4f wc = *(const v4f*)(sF1 + rowc + x0);
        v4f wu = *(const v4f*)(sF1 + rowm + x0);
        v4f wd = *(const v4f*)(sF1 + rowp + x0);
        float wls = sF1[rowc + xl], wrs = sF1[rowc + xr];
        // ---- b ----
        v4f bc = *(const v4f*)(sF2 + rowc + x0);
        v4f bu = *(const v4f*)(sF2 + rowm + x0);
        v4f bd = *(const v4f*)(sF2 + rowp + x0);
        float bls = sF2[rowc + xl], brs = sF2[rowc + xr];

        // Shifted vectors (replicate-clamped at image edges)
        v4f hL = (v4f){hls, hc.x, hc.y, hc.z};
        v4f hR = (v4f){hc.y, hc.z, hc.w, hrs};
        v4f wL = (v4f){wls, wc.x, wc.y, wc.z};
        v4f wR = (v4f){wc.y, wc.z, wc.w, wrs};
        v4f bL = (v4f){bls, bc.x, bc.y, bc.z};
        v4f bR = (v4f){bc.y, bc.z, bc.w, brs};

        // 5-point Laplacians (edge/replicate padding == clamped neighbors)
        v4f lapH = hu + hd + hL + hR - 4.0f * hc;
        v4f lapW = wu + wd + wL + wR - 4.0f * wc;
        v4f lapB = bu + bd + bL + bR - 4.0f * bc;

        // Gradients of h (reflect padding => exactly 0 at the boundary)
        v4f gx = hR - hL;
        if (x0 == 0)       gx.x = 0.0f;   // x == 0
        if (x0 + 4 == WW)  gx.w = 0.0f;   // x == W-1
        const float gyMask = (gy == 0 || gy == HH - 1) ? 0.0f : 1.0f;
        v4f gyv = (hu - hd) * gyMask;     // x[y-1] - x[y+1]

        // Reaction terms (Q == 1 so b**Q == b)
        v4f infil = 0.3f * hc * bc;                       // INFILT_RATE * h * b
        v4f uptak = 0.5f * wc * bc * (1.0f + 3.0f * bc);  // UPTAKE * w * b * (1 + ETA*b)
        v4f adv   = gx * dxv[r] + gyv * dyv[r];

        v4f dh = (0.1f  * lapH - 0.1f  * hc + 0.545f - infil - 0.2f * adv) * 0.1f;
        v4f dw = (0.05f * lapW - 0.2f  * wc + infil - uptak)               * 0.1f;
        v4f db = (0.05f * lapB - 0.15f * bc + 0.449f * uptak)              * 0.1f;

        // Streaming outputs: nontemporal stores keep L2 free for halo reuse
        const size_t gidx = tileBase + (size_t)r * WW;
        __builtin_nontemporal_store(hc + dh, (v4f*)(outH + gidx));
        __builtin_nontemporal_store(wc + dw, (v4f*)(outW + gidx));
        __builtin_nontemporal_store(bc + db, (v4f*)(outB + gidx));
    }
}

extern "C" void kernel_launch(void* const* d_in, const int* in_sizes, int n_in,
                              void* d_out, int out_size, void* d_ws, size_t ws_size,
                              hipStream_t stream) {
    const float* hp  = (const float*)d_in[0];
    const float* wp  = (const float*)d_in[1];
    const float* bp  = (const float*)d_in[2];
    const float* dxp = (const float*)d_in[3];
    const float* dyp = (const float*)d_in[4];
    float* out = (float*)d_out;

    const int Bn = in_sizes[0] / (HH * WW);  // batch (16)
    const size_t shmem = (size_t)3 * LROWS * WW * sizeof(float);  // 120 KB

    // Allow >default dynamic LDS (WGP supports up to 320 KB per workgroup).
    (void)hipFuncSetAttribute((const void*)nca_step_kernel,
                              hipFuncAttributeMaxDynamicSharedMemorySize,
                              (int)shmem);

    dim3 grid(Bn * (HH / TILE_H));
    nca_step_kernel<<<grid, TPB, shmem, stream>>>(hp, wp, bp, dxp, dyp, out, Bn);
}